// ASAPooling_25125558681611
// MI455X (gfx1250) — compile-verified
//
#include <hip/hip_runtime.h>
#include <stdint.h>

typedef float v2f __attribute__((ext_vector_type(2)));
typedef float v8f __attribute__((ext_vector_type(8)));

#define NEG_SLOPE 0.2f
#define SORT_N 4096

// ---------- helpers ----------

// Monotonic float<->uint encoding: order over encoded uints == order over floats.
__device__ __forceinline__ unsigned fenc(float f) {
  unsigned u = __float_as_uint(f);
  return u ^ ((u >> 31) ? 0xFFFFFFFFu : 0x80000000u);
}
__device__ __forceinline__ float fdec(unsigned u) {
  unsigned v = u ^ ((u & 0x80000000u) ? 0x80000000u : 0xFFFFFFFFu);
  return __uint_as_float(v);
}

__device__ __forceinline__ float wave_sum(float v) {
  for (int o = 16; o > 0; o >>= 1) v += __shfl_xor(v, o, 32);
  return v;
}

// ---------- WMMA fp32 GEMM:  C[M,Nd] = op(A)[M,Kd] * B[Kd,Nd] (+bias[col]) ----------
// Block = 128 threads = 4 waves. Block tile: 128(M) x 64(N).
// Each wave: 32x64 strip = 2 M-tiles x 4 N-tiles (8 v8f accumulators).
// B panel (16 x 64) staged in LDS once per block, shared by all 4 waves.
// REQUIRES: M % 128 == 0, Nd % 64 == 0, Kd % 16 == 0 (true for all uses here).
//
// V_WMMA_F32_16X16X4_F32 layouts (ISA 7.12.2):
//   A 16x4 : lanes 0-15 M=lane {v0=K0,v1=K1}; lanes 16-31 M=lane-16 {v0=K2,v1=K3}
//   B 4x16 : lanes 0-15 N=lane {v0=K0,v1=K1}; lanes 16-31 N=lane-16 {v0=K2,v1=K3}
//   C 16x16: vgpr r, lanes 0-15 -> (M=r, N=lane); lanes 16-31 -> (M=r+8, N=lane-16)
template <bool TA>
__global__ __launch_bounds__(128) void wmma_gemm_f32(
    const float* __restrict__ A, const float* __restrict__ B,
    float* __restrict__ C, const float* __restrict__ bias,
    int M, int Nd, int Kd, int lda, int ldb, int ldc) {
  __shared__ float Bs[16 * 64];
  const int tid  = threadIdx.x;       // 0..127
  const int lane = tid & 31;
  const int wave = tid >> 5;          // 0..3
  const int tn0  = blockIdx.x * 64;
  const int tm   = blockIdx.y * 128 + wave * 32;

  v8f acc[8];                         // [mi*4 + ni]
#pragma unroll
  for (int i = 0; i < 8; ++i) acc[i] = (v8f){};

  const int r15   = lane & 15;
  const int hk    = (lane >> 4) << 1; // 0 or 2
  const int rowA0 = tm + r15;
  const int rowA1 = tm + 16 + r15;

  for (int k0 = 0; k0 < Kd; k0 += 16) {
    __syncthreads();                  // protect previous chunk's reads
#pragma unroll
    for (int i = 0; i < 8; ++i) {     // 16*64 / 128 threads = 8 each
      int idx = tid + i * 128;
      int kr = idx >> 6, cc = idx & 63;
      Bs[idx] = B[(size_t)(k0 + kr) * ldb + tn0 + cc];
    }
    __syncthreads();

#pragma unroll
    for (int kk = 0; kk < 16; kk += 4) {
      v2f a0, a1;
      if (TA) {
        const float* Ac = A + (size_t)(k0 + kk + hk) * lda;
        a0.x = Ac[rowA0];       a1.x = Ac[rowA1];
        a0.y = Ac[lda + rowA0]; a1.y = Ac[lda + rowA1];
      } else {
        const float* Ar0 = A + (size_t)rowA0 * lda + (k0 + kk + hk);
        const float* Ar1 = A + (size_t)rowA1 * lda + (k0 + kk + hk);
        a0.x = Ar0[0]; a0.y = Ar0[1];
        a1.x = Ar1[0]; a1.y = Ar1[1];
      }
      const float* b0p = Bs + (kk + hk) * 64 + r15;
      const float* b1p = Bs + (kk + hk + 1) * 64 + r15;
#pragma unroll
      for (int ni = 0; ni < 4; ++ni) {
        v2f b;
        b.x = b0p[ni * 16];
        b.y = b1p[ni * 16];
        acc[ni]     = __builtin_amdgcn_wmma_f32_16x16x4_f32(false, a0, false, b, (short)0, acc[ni],     false, false);
        acc[4 + ni] = __builtin_amdgcn_wmma_f32_16x16x4_f32(false, a1, false, b, (short)0, acc[4 + ni], false, false);
      }
    }
  }

  const int mrow0 = tm + (lane >> 4) * 8;
#pragma unroll
  for (int mi = 0; mi < 2; ++mi) {
#pragma unroll
    for (int ni = 0; ni < 4; ++ni) {
      int col = tn0 + ni * 16 + r15;
      float bv = bias ? bias[col] : 0.0f;
      v8f av = acc[mi * 4 + ni];
#pragma unroll
      for (int r = 0; r < 8; ++r)
        C[(size_t)(mrow0 + mi * 16 + r) * ldc + col] = av[r] + bv;
    }
  }
}

// ---------- per-edge / per-node kernels ----------

// x_q[d, :] = max over incoming edges of x[src, :]   (encoded-uint atomicMax)
__global__ __launch_bounds__(256) void seg_max_kernel(
    const float* __restrict__ x, const int* __restrict__ ei,
    unsigned* __restrict__ xq_enc, int E, int Et, int D) {
  long long idx = (long long)blockIdx.x * blockDim.x + threadIdx.x;
  long long total = (long long)Et * D;
  if (idx >= total) return;
  int e  = (int)(idx / D);
  int dd = (int)(idx % D);
  int s, d;
  if (e < E) { s = ei[e]; d = ei[E + e]; } else { s = d = e - E; }
  atomicMax(xq_enc + (size_t)d * D + dd, fenc(x[(size_t)s * D + dd]));
}

__global__ __launch_bounds__(256) void decode_kernel(unsigned* __restrict__ buf, long long n) {
  long long i = (long long)blockIdx.x * blockDim.x + threadIdx.x;
  if (i < n) buf[i] = __float_as_uint(fdec(buf[i]));
}

// score_e = dot(xlin[dst], att_w[:D]) + dot(x[src], att_w[D:]) + att_b ; leaky_relu ; seg-max
__global__ __launch_bounds__(256) void edge_score_kernel(
    const float* __restrict__ xlin, const float* __restrict__ x,
    const int* __restrict__ ei, const float* __restrict__ att_w,
    const float* __restrict__ att_b, float* __restrict__ score,
    unsigned* __restrict__ menc, int E, int Et, int D) {
  int e = blockIdx.x * 8 + (threadIdx.x >> 5);
  if (e >= Et) return;
  int lane = threadIdx.x & 31;
  int s, d;
  if (e < E) { s = ei[e]; d = ei[E + e]; } else { s = d = e - E; }
  float acc = 0.f;
  for (int dd = lane; dd < D; dd += 32)
    acc += xlin[(size_t)d * D + dd] * att_w[dd] + x[(size_t)s * D + dd] * att_w[D + dd];
  acc = wave_sum(acc);
  if (lane == 0) {
    acc += att_b[0];
    acc = acc >= 0.f ? acc : NEG_SLOPE * acc;
    score[e] = acc;
    atomicMax(menc + d, fenc(acc));
  }
}

__global__ __launch_bounds__(256) void edge_exp_kernel(
    const float* __restrict__ score, const int* __restrict__ ei,
    const unsigned* __restrict__ menc, float* __restrict__ expv,
    float* __restrict__ ssum, int E, int Et) {
  int e = blockIdx.x * blockDim.x + threadIdx.x;
  if (e >= Et) return;
  int d = (e < E) ? ei[E + e] : (e - E);
  float ex = __expf(score[e] - fdec(menc[d]));
  expv[e] = ex;
  atomicAdd(ssum + d, ex);
}

__global__ __launch_bounds__(256) void edge_sm_kernel(
    const float* __restrict__ expv, const int* __restrict__ ei,
    const float* __restrict__ ssum, float* __restrict__ sm, int E, int Et) {
  int e = blockIdx.x * blockDim.x + threadIdx.x;
  if (e >= Et) return;
  int d = (e < E) ? ei[E + e] : (e - E);
  sm[e] = expv[e] / (ssum[d] + 1e-16f);
}

// x_new[dst,:] += sm_e * x[src,:]
__global__ __launch_bounds__(256) void xnew_scatter_kernel(
    const float* __restrict__ x, const int* __restrict__ ei,
    const float* __restrict__ sm, float* __restrict__ xnew,
    int E, int Et, int D) {
  long long idx = (long long)blockIdx.x * blockDim.x + threadIdx.x;
  long long total = (long long)Et * D;
  if (idx >= total) return;
  int e  = (int)(idx / D);
  int dd = (int)(idx % D);
  int s, d;
  if (e < E) { s = ei[e]; d = ei[E + e]; } else { s = d = e - E; }
  atomicAdd(xnew + (size_t)d * D + dd, x[(size_t)s * D + dd] * sm[e]);
}

// a = x_new@w1+b1 ; b = x_new@w2 ; fitpre = x_new@w3+b3
__global__ __launch_bounds__(256) void node_dots_kernel(
    const float* __restrict__ xnew, const float* __restrict__ w1,
    const float* __restrict__ b1, const float* __restrict__ w2,
    const float* __restrict__ w3, const float* __restrict__ b3,
    float* __restrict__ aN, float* __restrict__ bN,
    float* __restrict__ fitpre, int N, int D) {
  int n = blockIdx.x * 8 + (threadIdx.x >> 5);
  if (n >= N) return;
  int lane = threadIdx.x & 31;
  float s1 = 0.f, s2 = 0.f, s3 = 0.f;
  for (int dd = lane; dd < D; dd += 32) {
    float v = xnew[(size_t)n * D + dd];
    s1 += v * w1[dd]; s2 += v * w2[dd]; s3 += v * w3[dd];
  }
  s1 = wave_sum(s1); s2 = wave_sum(s2); s3 = wave_sum(s3);
  if (lane == 0) { aN[n] = s1 + b1[0]; bN[n] = s2; fitpre[n] = s3 + b3[0]; }
}

__global__ __launch_bounds__(256) void edge_fit_kernel(
    const int* __restrict__ ei, const float* __restrict__ aN,
    const float* __restrict__ bN, float* __restrict__ fitpre, int E, int Et) {
  int e = blockIdx.x * blockDim.x + threadIdx.x;
  if (e >= Et) return;
  int s, d;
  if (e < E) { s = ei[e]; d = ei[E + e]; } else { s = d = e - E; }
  atomicAdd(fitpre + d, aN[s] - bN[d]);
}

// fit = sigmoid(fitpre); pack descending sort key: (enc(fit)<<32)|~idx  (ties -> lower idx first)
__global__ __launch_bounds__(256) void fit_keys_kernel(
    const float* __restrict__ fitpre, float* __restrict__ fit,
    unsigned long long* __restrict__ keys, int N) {
  int n = blockIdx.x * blockDim.x + threadIdx.x;
  if (n >= N) return;
  float f = 1.0f / (1.0f + __expf(-fitpre[n]));
  fit[n] = f;
  keys[n] = ((unsigned long long)fenc(f) << 32) | (unsigned)(~(unsigned)n);
}

// single-workgroup bitonic sort (descending) of SORT_N u64 keys; emits perm, fv, inv
__global__ __launch_bounds__(1024) void topk_sort_kernel(
    const unsigned long long* __restrict__ keys, const float* __restrict__ fit,
    int* __restrict__ perm, float* __restrict__ fv, int* __restrict__ inv,
    int N, int K) {
  __shared__ unsigned long long sk[SORT_N];
  const int tid = threadIdx.x;
  for (int i = tid; i < SORT_N; i += 1024) sk[i] = (i < N) ? keys[i] : 0ull;
  for (int i = tid; i < N; i += 1024) inv[i] = K;
  __syncthreads();
  for (int size = 2; size <= SORT_N; size <<= 1) {
    for (int stride = size >> 1; stride > 0; stride >>= 1) {
      for (int i = tid; i < SORT_N; i += 1024) {
        int j = i ^ stride;
        if (j > i) {
          unsigned long long ki = sk[i], kj = sk[j];
          bool descBlock = ((i & size) == 0);
          bool doSwap = descBlock ? (ki < kj) : (ki > kj);
          if (doSwap) { sk[i] = kj; sk[j] = ki; }
        }
      }
      __syncthreads();
    }
  }
  for (int i = tid; i < K; i += 1024) {
    unsigned idx = ~(unsigned)(sk[i] & 0xFFFFFFFFull);
    perm[i] = (int)idx;
    fv[i] = fit[idx];
  }
  __syncthreads();
  for (int i = tid; i < K; i += 1024) inv[perm[i]] = i;
}

__global__ __launch_bounds__(256) void xout_kernel(
    const float* __restrict__ xnew, const int* __restrict__ perm,
    const float* __restrict__ fv, float* __restrict__ xout, int K, int D) {
  long long idx = (long long)blockIdx.x * blockDim.x + threadIdx.x;
  if (idx >= (long long)K * D) return;
  int i = (int)(idx / D), dd = (int)(idx % D);
  xout[idx] = xnew[(size_t)perm[i] * D + dd] * fv[i];
}

// Ad[src,dst] += w ; Sd[src, inv[dst]] += sm (skip dropped column)
__global__ __launch_bounds__(256) void build_SA_kernel(
    const int* __restrict__ ei, const float* __restrict__ ew,
    const float* __restrict__ sm, const int* __restrict__ inv,
    float* __restrict__ Ad, float* __restrict__ Sd,
    int E, int Et, int N, int K) {
  int e = blockIdx.x * blockDim.x + threadIdx.x;
  if (e >= Et) return;
  int s, d; float w;
  if (e < E) { s = ei[e]; d = ei[E + e]; w = ew[e]; } else { s = d = e - E; w = 1.0f; }
  atomicAdd(Ad + (size_t)s * N + d, w);
  int col = inv[d];
  if (col < K) atomicAdd(Sd + (size_t)s * K + col, sm[e]);
}

__global__ __launch_bounds__(256) void diag_zero_kernel(float* __restrict__ Apool, int K) {
  int i = blockIdx.x * blockDim.x + threadIdx.x;
  if (i < K) Apool[(size_t)i * K + i] = 0.0f;
}

// ---------- launcher ----------

extern "C" void kernel_launch(void* const* d_in, const int* in_sizes, int n_in,
                              void* d_out, int out_size, void* d_ws, size_t ws_size,
                              hipStream_t stream) {
  const float* x     = (const float*)d_in[0];
  const int*   ei    = (const int*)d_in[2];
  const float* ew    = (const float*)d_in[3];
  const float* lin_W = (const float*)d_in[4];
  const float* lin_b = (const float*)d_in[5];
  const float* att_w = (const float*)d_in[6];
  const float* att_b = (const float*)d_in[7];
  const float* w1    = (const float*)d_in[8];
  const float* b1    = (const float*)d_in[9];
  const float* w2    = (const float*)d_in[10];
  const float* w3    = (const float*)d_in[11];
  const float* b3    = (const float*)d_in[12];

  const int N  = in_sizes[1];          // 4096
  const int D  = in_sizes[5];          // 256
  const int E  = in_sizes[3];          // 98304
  const int Et = E + N;                // + self loops
  const int K  = (N + 1) / 2;          // ceil(0.5*N) = 2048

  // workspace carve-out
  char* base = (char*)d_ws;
  size_t off = 0;
  auto carve = [&](size_t bytes) -> char* {
    char* p = base + off;
    off = (off + bytes + 255) & ~(size_t)255;
    return p;
  };
  float*    Ad     = (float*)carve((size_t)N * N * 4);
  float*    Sd     = (float*)carve((size_t)N * K * 4);
  float*    T      = (float*)carve((size_t)N * K * 4);
  unsigned* xq     = (unsigned*)carve((size_t)N * D * 4);  // enc -> decoded float in place
  float*    xlin   = (float*)carve((size_t)N * D * 4);
  float*    xnew   = (float*)carve((size_t)N * D * 4);
  float*    score  = (float*)carve((size_t)Et * 4);
  float*    expv   = (float*)carve((size_t)Et * 4);
  float*    sm     = (float*)carve((size_t)Et * 4);
  unsigned* menc   = (unsigned*)carve((size_t)N * 4);
  float*    ssum   = (float*)carve((size_t)N * 4);
  float*    aN     = (float*)carve((size_t)N * 4);
  float*    bN     = (float*)carve((size_t)N * 4);
  float*    fitpre = (float*)carve((size_t)N * 4);
  float*    fit    = (float*)carve((size_t)N * 4);
  unsigned long long* keys = (unsigned long long*)carve((size_t)N * 8);
  int*      perm   = (int*)carve((size_t)K * 4);
  float*    fv     = (float*)carve((size_t)K * 4);
  int*      inv    = (int*)carve((size_t)N * 4);

  float* xout  = (float*)d_out;                 // [K, D]
  float* Apool = (float*)d_out + (size_t)K * D; // [K, K]

  // zero-init accumulators (0x00000000 is the identity for the encoded umax too)
  hipMemsetAsync(Ad,   0, (size_t)N * N * 4, stream);
  hipMemsetAsync(Sd,   0, (size_t)N * K * 4, stream);
  hipMemsetAsync(xq,   0, (size_t)N * D * 4, stream);
  hipMemsetAsync(xnew, 0, (size_t)N * D * 4, stream);
  hipMemsetAsync(menc, 0, (size_t)N * 4, stream);
  hipMemsetAsync(ssum, 0, (size_t)N * 4, stream);

  // 1) x_q = segment_max(x[src], dst)
  {
    long long tot = (long long)Et * D;
    seg_max_kernel<<<(unsigned)((tot + 255) / 256), 256, 0, stream>>>(x, ei, xq, E, Et, D);
    long long nd = (long long)N * D;
    decode_kernel<<<(unsigned)((nd + 255) / 256), 256, 0, stream>>>(xq, nd);
  }
  // 2) xlin = x_q @ lin_W + lin_b   (WMMA fp32, LDS-staged B)
  wmma_gemm_f32<false><<<dim3(D / 64, N / 128), 128, 0, stream>>>(
      (const float*)xq, lin_W, xlin, lin_b, N, D, D, D, D, D);
  // 3) per-edge attention score + leaky_relu + per-dst max
  edge_score_kernel<<<(Et + 7) / 8, 256, 0, stream>>>(xlin, x, ei, att_w, att_b, score, menc, E, Et, D);
  // 4) scatter softmax
  edge_exp_kernel<<<(Et + 255) / 256, 256, 0, stream>>>(score, ei, menc, expv, ssum, E, Et);
  edge_sm_kernel<<<(Et + 255) / 256, 256, 0, stream>>>(expv, ei, ssum, sm, E, Et);
  // 5) x_new = segment_sum(sm * x[src])
  {
    long long tot = (long long)Et * D;
    xnew_scatter_kernel<<<(unsigned)((tot + 255) / 256), 256, 0, stream>>>(x, ei, sm, xnew, E, Et, D);
  }
  // 6) LEConv fitness
  node_dots_kernel<<<(N + 7) / 8, 256, 0, stream>>>(xnew, w1, b1, w2, w3, b3, aN, bN, fitpre, N, D);
  edge_fit_kernel<<<(Et + 255) / 256, 256, 0, stream>>>(ei, aN, bN, fitpre, E, Et);
  fit_keys_kernel<<<(N + 255) / 256, 256, 0, stream>>>(fitpre, fit, keys, N);
  // 7) exact top-k (descending value, ascending index on ties)
  topk_sort_kernel<<<1, 1024, 0, stream>>>(keys, fit, perm, fv, inv, N, K);
  // 8) x_out = x_new[perm] * fv
  {
    long long tot = (long long)K * D;
    xout_kernel<<<(unsigned)((tot + 255) / 256), 256, 0, stream>>>(xnew, perm, fv, xout, K, D);
  }
  // 9) densify A and S, then Apool = Sd^T @ (Ad @ Sd) via two WMMA GEMMs
  build_SA_kernel<<<(Et + 255) / 256, 256, 0, stream>>>(ei, ew, sm, inv, Ad, Sd, E, Et, N, K);
  wmma_gemm_f32<false><<<dim3(K / 64, N / 128), 128, 0, stream>>>(
      Ad, Sd, T, nullptr, N, K, N, N, K, K);
  wmma_gemm_f32<true><<<dim3(K / 64, K / 128), 128, 0, stream>>>(
      Sd, T, Apool, nullptr, K, K, N, K, K, K);
  // 10) remove_diag
  diag_zero_kernel<<<(K + 255) / 256, 256, 0, stream>>>(Apool, K);
}